// DyGCN_14130442404424
// MI455X (gfx1250) — compile-verified
//
#include <hip/hip_runtime.h>
#include <hip/hip_bf16.h>

// Problem constants (reference: B,N,E,K,DI,DO = 16,1024,16,3,32,32)
#define Bsz 16
#define Nn  1024
#define Ee  16
#define DIc 32
#define DOc 32

#define GAS __attribute__((address_space(1)))
#define LAS __attribute__((address_space(3)))

typedef float v2f __attribute__((ext_vector_type(2)));
typedef float v8f __attribute__((ext_vector_type(8)));
typedef int   v4i __attribute__((ext_vector_type(4)));

// V_WMMA_F32_16X16X4_F32 : D(16x16 f32) = A(16x4 f32) * B(4x16 f32) + C
// Layout (ISA 7.12.2): A frag lane l: row m = l&15, half h = l>>4,
//   vgpr0 = A[m][2h], vgpr1 = A[m][2h+1].
// B frag: col n = l&15, vgpr0 = B[2h][n], vgpr1 = B[2h+1][n].
// C/D frag: vgpr r, lane l -> row m = r + 8h, col n = l&15.
__device__ __forceinline__ v8f wmma_f32(v2f a, v2f b, v8f c) {
    return __builtin_amdgcn_wmma_f32_16x16x4_f32(
        false, a, false, b, (short)0, c, false, false);
}

// ---------------------------------------------------------------------------
// Kernel 1: S = softmax(relu(all_emb @ all_emb^T), axis=-1)   per batch.
// One block = one 16-row strip; 8 waves compute 64 16x16 WMMA tiles (K=16 as
// four K=4 slices), strip staged in 64KB LDS, then row softmax.
// Column-tile loop has a uniform literal trip count (8) so no EXEC masking is
// generated around the WMMA body (EXEC must be all-ones for WMMA).
// ---------------------------------------------------------------------------
__global__ __launch_bounds__(256) void adj_softmax_kernel(
    const float* __restrict__ emb, float* __restrict__ S) {
    __shared__ float tile[16][Nn];                       // 64 KB
    const int b    = blockIdx.x >> 6;                    // / (N/16)
    const int row0 = (blockIdx.x & 63) << 4;
    const int tid  = threadIdx.x;
    const int wave = tid >> 5;
    const int lane = tid & 31;
    const int h    = lane >> 4;
    const int m16  = lane & 15;

    // A fragments: rows (row0+m) of emb, K = 0..15 in four slices of 4.
    const float* rowp = emb + ((size_t)b * Nn + row0 + m16) * Ee;
    v2f aS[4];
#pragma unroll
    for (int s = 0; s < 4; ++s)
        aS[s] = *(const v2f*)(rowp + 4 * s + 2 * h);

#pragma unroll
    for (int it = 0; it < 8; ++it) {                     // uniform trip count
        const int col0 = (it * 8 + wave) << 4;
        const float* colp = emb + ((size_t)b * Nn + col0 + m16) * Ee;
        v8f acc = {};
#pragma unroll
        for (int s = 0; s < 4; ++s) {
            v2f bf = *(const v2f*)(colp + 4 * s + 2 * h);  // B[k][n]=emb[n][k]
            acc = wmma_f32(aS[s], bf, acc);
        }
#pragma unroll
        for (int r = 0; r < 8; ++r)                       // fused ReLU
            tile[r + 8 * h][col0 + m16] = fmaxf(acc[r], 0.0f);
    }
    __syncthreads();

    // Row softmax: each row owned by one 16-lane half-wave (thread t -> row
    // t/16, column phase t%16), reductions via __shfl_xor within width 16.
    const int row = tid >> 4;
    const int c0  = tid & 15;
    float mx = -3.402823466e38f;
#pragma unroll 4
    for (int j = 0; j < 64; ++j)
        mx = fmaxf(mx, tile[row][c0 + 16 * j]);
#pragma unroll
    for (int off = 8; off >= 1; off >>= 1)
        mx = fmaxf(mx, __shfl_xor(mx, off, 16));
    float sm = 0.0f;
#pragma unroll 4
    for (int j = 0; j < 64; ++j) {
        float e = __expf(tile[row][c0 + 16 * j] - mx);
        tile[row][c0 + 16 * j] = e;
        sm += e;
    }
#pragma unroll
    for (int off = 8; off >= 1; off >>= 1)
        sm += __shfl_xor(sm, off, 16);
    const float inv = 1.0f / sm;
    float* srow = S + ((size_t)b * Nn + row0 + row) * Nn;
#pragma unroll 4
    for (int j = 0; j < 64; ++j)
        srow[c0 + 16 * j] = tile[row][c0 + 16 * j] * inv;
}

// ---------------------------------------------------------------------------
// Kernel 2: Y = alpha * (S @ X) + beta * Z     (per batch; X,Z,Y are [N,32])
// 4 waves/block, each owns a 16-row tile; X streamed through LDS in 128-row
// chunks using gfx1250 GLOBAL_LOAD_ASYNC_TO_LDS_B128 (ASYNCcnt) when the
// builtin is available, else a float4 VGPR round-trip.
// Chebyshev recursion applied to x only:
//   call 1: Y1 = 1*S@x            (= T1 x)
//   call 2: Y2 = 2*S@Y1 - 1*x     (= T2 x = (2 S S - I) x)
// ---------------------------------------------------------------------------
template <bool HAS_BETA>
__global__ __launch_bounds__(128) void propagate_kernel(
    const float* __restrict__ S, const float* __restrict__ X,
    const float* __restrict__ Z, float* __restrict__ Y,
    float alpha, float beta) {
    __shared__ float ldsX[128][DIc];                     // 16 KB
    const int b     = blockIdx.x >> 4;
    const int tile4 = (blockIdx.x & 15) << 2;
    const int tid   = threadIdx.x;
    const int wave  = tid >> 5;
    const int lane  = tid & 31;
    const int h     = lane >> 4;
    const int m16   = lane & 15;
    const int row0  = (tile4 + wave) * 16;

    const float* srow = S + ((size_t)b * Nn + row0 + m16) * Nn;
    const float* xb   = X + (size_t)b * Nn * DIc;
    v8f acc0 = {}, acc1 = {};

    for (int kc = 0; kc < Nn; kc += 128) {
        __syncthreads();
        const float4* src = (const float4*)(xb + (size_t)kc * DIc);
        float4*       dst = (float4*)(&ldsX[0][0]);
#if __has_builtin(__builtin_amdgcn_global_load_async_to_lds_b128)
        // Async DMA memory->LDS, no VGPR round trip; tracked by ASYNCcnt.
        // Param 0 is a global (addrspace(1)) int4*, per clang's signature.
#pragma unroll
        for (int it = 0; it < (128 * DIc / 4) / 128; ++it) {
            const int idx = it * 128 + tid;
            __builtin_amdgcn_global_load_async_to_lds_b128(
                (GAS v4i*)(src + idx), (LAS v4i*)(dst + idx), 0, 0);
        }
        asm volatile("s_wait_asynccnt 0x0" ::: "memory");
#else
        for (int idx = tid; idx < 128 * DIc / 4; idx += 128)
            dst[idx] = src[idx];
#endif
        __syncthreads();
        if (kc + 128 < Nn)                                // hint next S chunk
            __builtin_prefetch(srow + kc + 128, 0, 0);
#pragma unroll 4
        for (int kk = 0; kk < 128; kk += 4) {
            v2f a = *(const v2f*)(srow + kc + kk + 2 * h);
            v2f b0, b1;
            b0.x = ldsX[kk + 2 * h][m16];
            b0.y = ldsX[kk + 2 * h + 1][m16];
            b1.x = ldsX[kk + 2 * h][m16 + 16];
            b1.y = ldsX[kk + 2 * h + 1][m16 + 16];
            acc0 = wmma_f32(a, b0, acc0);
            acc1 = wmma_f32(a, b1, acc1);
        }
    }
#pragma unroll
    for (int r = 0; r < 8; ++r) {
        const int m = r + 8 * h;
        float* yrow = Y + ((size_t)b * Nn + row0 + m) * DIc;
        if (HAS_BETA) {
            const float* zrow = Z + ((size_t)b * Nn + row0 + m) * DIc;
            yrow[m16]      = alpha * acc0[r] + beta * zrow[m16];
            yrow[m16 + 16] = alpha * acc1[r] + beta * zrow[m16 + 16];
        } else {
            yrow[m16]      = alpha * acc0[r];
            yrow[m16 + 16] = alpha * acc1[r];
        }
    }
}

// ---------------------------------------------------------------------------
// Kernel 3: out[b,n,o] = sum_{e,k,i} stock[b,n,e]*x_g[b,k,n,i]*W[e,k,i,o]
//                        + sum_e stock[b,n,e]*bias[e,o]
// As one WMMA GEMM with K' = E*K*DI + E = 1552:
//   A row m, col kk<1536:  u = stock[m][kk/96] * xg[m][kk%96]   (on the fly)
//   A row m, col kk>=1536: u = stock[m][kk-1536]                (bias rows)
//   B row kk: weights_pool is (E,K,DI,DO) so flat row kk is exactly W+kk*32;
//             bias rows come from bias_pool.
// ---------------------------------------------------------------------------
__global__ __launch_bounds__(128) void output_kernel(
    const float* __restrict__ X,  const float* __restrict__ Y1,
    const float* __restrict__ Y2, const float* __restrict__ stock,
    const float* __restrict__ Wp, const float* __restrict__ biasp,
    float* __restrict__ out) {
    __shared__ float stockT[4][16][Ee];                  // 4 KB
    __shared__ float xgT[4][16][96];                     // 24 KB
    const int tid    = threadIdx.x;
    const int wave   = tid >> 5;
    const int lane   = tid & 31;
    const int h      = lane >> 4;
    const int m16    = lane & 15;
    const int tileId = blockIdx.x * 4 + wave;            // 0..1023
    const int b      = tileId >> 6;
    const int node0  = (tileId & 63) << 4;
    const size_t base = (size_t)b * Nn + node0;

#pragma unroll
    for (int it = 0; it < 8; ++it) {                     // 16*Ee / 32
        const int idx = it * 32 + lane;
        const int m = idx >> 4, e = idx & 15;
        stockT[wave][m][e] = stock[(base + m) * Ee + e];
    }
#pragma unroll
    for (int it = 0; it < 48; ++it) {                    // 16*96 / 32
        const int idx = it * 32 + lane;
        const int m = idx / 96, f = idx - m * 96;
        const int k = f >> 5, i = f & 31;
        const float* src = (k == 0) ? X : (k == 1) ? Y1 : Y2;
        xgT[wave][m][f] = src[(base + m) * DIc + i];
    }
    __syncthreads();

    v8f acc0 = {}, acc1 = {};
    const int KT = Ee * 96;                              // 1536
    for (int kk = 0; kk < KT + Ee; kk += 4) {            // 388 k-steps
        const int k0 = kk + 2 * h;
        v2f a;
        if (kk < KT) {                                   // uniform branch
            const int e0 = k0 / 96,       r0 = k0 - e0 * 96;
            const int e1 = (k0 + 1) / 96, r1 = (k0 + 1) - e1 * 96;
            a.x = stockT[wave][m16][e0] * xgT[wave][m16][r0];
            a.y = stockT[wave][m16][e1] * xgT[wave][m16][r1];
        } else {                                         // bias rows
            a.x = stockT[wave][m16][k0 - KT];
            a.y = stockT[wave][m16][k0 - KT + 1];
        }
        const float* wrow = (kk < KT) ? (Wp + (size_t)kk * DOc)
                                      : (biasp + (size_t)(kk - KT) * DOc);
        v2f b0, b1;
        b0.x = wrow[(2 * h) * DOc + m16];
        b0.y = wrow[(2 * h + 1) * DOc + m16];
        b1.x = wrow[(2 * h) * DOc + m16 + 16];
        b1.y = wrow[(2 * h + 1) * DOc + m16 + 16];
        acc0 = wmma_f32(a, b0, acc0);
        acc1 = wmma_f32(a, b1, acc1);
    }
#pragma unroll
    for (int r = 0; r < 8; ++r) {
        const int m = r + 8 * h;
        float* orow = out + (base + m) * DOc;
        orow[m16]      = acc0[r];
        orow[m16 + 16] = acc1[r];
    }
}

// ---------------------------------------------------------------------------
// Host launcher. Workspace layout (fp32):
//   S  : B*N*N      = 16M floats (64 MB)   at d_ws + 0
//   Y1 : B*N*DI     = 512K floats (2 MB)
//   Y2 : B*N*DI     = 512K floats (2 MB)
// ---------------------------------------------------------------------------
extern "C" void kernel_launch(void* const* d_in, const int* in_sizes, int n_in,
                              void* d_out, int out_size, void* d_ws, size_t ws_size,
                              hipStream_t stream) {
    (void)in_sizes; (void)n_in; (void)out_size; (void)ws_size;
    const float* x       = (const float*)d_in[0];
    const float* all_emb = (const float*)d_in[1];
    const float* stock   = (const float*)d_in[2];
    const float* Wp      = (const float*)d_in[3];
    const float* biasp   = (const float*)d_in[4];
    float* out = (float*)d_out;

    float* S  = (float*)d_ws;
    float* Y1 = S  + (size_t)Bsz * Nn * Nn;
    float* Y2 = Y1 + (size_t)Bsz * Nn * DIc;

    // 1) S = softmax(relu(E E^T))        [1024 blocks x 256 thr]
    adj_softmax_kernel<<<Bsz * (Nn / 16), 256, 0, stream>>>(all_emb, S);
    // 2) Y1 = S @ x                      [256 blocks x 128 thr]
    propagate_kernel<false><<<Bsz * (Nn / 64), 128, 0, stream>>>(
        S, x, x, Y1, 1.0f, 0.0f);
    // 3) Y2 = 2 S @ Y1 - x  (= T2 x)
    propagate_kernel<true><<<Bsz * (Nn / 64), 128, 0, stream>>>(
        S, Y1, x, Y2, 2.0f, -1.0f);
    // 4) out = fused node-wise weighted conv + bias
    output_kernel<<<(Bsz * Nn / 16) / 4, 128, 0, stream>>>(x, Y1, Y2, stock, Wp,
                                                           biasp, out);
}